// IPA_61083024883810
// MI455X (gfx1250) — compile-verified
//
#include <hip/hip_runtime.h>
#include <math.h>

// ---------------------------------------------------------------------------
// IPA (Invariant Point Attention) for MI455X / gfx1250, wave32 + WMMA f32.
// Shapes: B=2, N=512, S=384, Z=128, H=12, C=16, PQ=4, PV=8.
//
// Roofline: z is 256MB and dominates all traffic (everything else is
// L2-resident). One workgroup per attention row (b,i) stages z[b,i,:,:]
// (256KB) ONCE into the 320KB LDS via the Tensor Data Mover (or async LDS
// loads), and reuses it for both z-GEMMs (bias = z@Wb and opair = a@z) with
// V_WMMA_F32_16X16X4_F32. Being HBM-bound, f32 WMMA costs nothing vs bf16.
// ---------------------------------------------------------------------------

typedef float v2f __attribute__((ext_vector_type(2)));
typedef float v8f __attribute__((ext_vector_type(8)));
typedef unsigned int u32x4 __attribute__((ext_vector_type(4)));
typedef int i32x4 __attribute__((ext_vector_type(4)));
typedef int i32x8 __attribute__((ext_vector_type(8)));

#if defined(__AMDGCN__) && __has_builtin(__builtin_amdgcn_tensor_load_to_lds) && \
    __has_builtin(__builtin_amdgcn_s_wait_tensorcnt)
#define IPA_HAS_TDM 1
#else
#define IPA_HAS_TDM 0
#endif

#if defined(__AMDGCN__) && __has_builtin(__builtin_amdgcn_global_load_async_to_lds_b128)
#define IPA_HAS_ASYNC 1
#else
#define IPA_HAS_ASYNC 0
#endif

#define H_  12
#define C_  16
#define PQ_ 4
#define PV_ 8
#define N_  512
#define Z_  128
#define ZP_ 132          // padded LDS row stride for z tile (bank-conflict free)
#define CATW_ 2112       // H*(C + 4*PV + Z)

__device__ __forceinline__ v8f wmma4(v2f a, v2f b, v8f c) {
  // D = A(16x4 f32) * B(4x16 f32) + C(16x16 f32)
  return __builtin_amdgcn_wmma_f32_16x16x4_f32(false, a, false, b, (short)0, c,
                                               false, false);
}

// ---------------------------------------------------------------------------
// Generic f32 WMMA GEMM: C[M,N] = A[M,K] @ B[K,N], row-major.
// Requires M,N multiples of 16 and K multiple of 4. One wave per 16x16 tile.
// ---------------------------------------------------------------------------
__global__ __launch_bounds__(32) void gemm_f32_wmma(
    const float* __restrict__ A, const float* __restrict__ B,
    float* __restrict__ C, int M, int N, int K) {
  int lane = threadIdx.x & 31;
  int lr = lane & 15, kh = lane >> 4;
  int m0 = blockIdx.x * 16, n0 = blockIdx.y * 16;
  const float* arow = A + (size_t)(m0 + lr) * K + 2 * kh;
  const float* bcol = B + (size_t)(2 * kh) * N + n0 + lr;
  v8f acc = {};
  for (int k0 = 0; k0 < K; k0 += 4) {
    v2f a, b;
    a.x = arow[k0];
    a.y = arow[k0 + 1];
    b.x = bcol[(size_t)k0 * N];
    b.y = bcol[(size_t)(k0 + 1) * N];
    acc = wmma4(a, b, acc);
  }
  float* crow = C + (size_t)(m0 + 8 * kh) * N + n0 + lr;
  for (int r = 0; r < 8; ++r) crow[(size_t)r * N] = acc[r];
}

// ---------------------------------------------------------------------------
// Map local-frame points to global: p <- R p + t   (in place)
// ---------------------------------------------------------------------------
__global__ __launch_bounds__(256) void point_to_global(
    float* __restrict__ p, const float* __restrict__ R,
    const float* __restrict__ t, int HP, int total) {
  int idx = blockIdx.x * 256 + threadIdx.x;
  if (idx >= total) return;
  int bn = idx / HP;
  float* pp = p + (size_t)idx * 3;
  float x = pp[0], y = pp[1], z = pp[2];
  const float* Rr = R + (size_t)bn * 9;
  const float* tt = t + (size_t)bn * 3;
  pp[0] = Rr[0] * x + Rr[1] * y + Rr[2] * z + tt[0];
  pp[1] = Rr[3] * x + Rr[4] * y + Rr[5] * z + tt[1];
  pp[2] = Rr[6] * x + Rr[7] * y + Rr[8] * z + tt[2];
}

// sq[bn,h] = sum_{p,x} pg[bn,h,p,x]^2 ; one thread per (bn,h)
__global__ __launch_bounds__(256) void point_sqsum(
    const float* __restrict__ pg, float* __restrict__ sq, int P, int total) {
  int idx = blockIdx.x * 256 + threadIdx.x;
  if (idx >= total) return;
  const float* pp = pg + (size_t)idx * P * 3;
  float s = 0.f;
  for (int e = 0; e < P * 3; ++e) s += pp[e] * pp[e];
  sq[idx] = s;
}

// ---------------------------------------------------------------------------
// Fused attention: one block per (b,i). 256 threads = 8 waves.
// LDS: z tile (512x132), logits/probs (16x512), padded Wb (128x16), row data.
// ---------------------------------------------------------------------------
__global__ __launch_bounds__(256) void ipa_attn(
    const float* __restrict__ q,   const float* __restrict__ k,
    const float* __restrict__ v,   const float* __restrict__ qpg,
    const float* __restrict__ kpg, const float* __restrict__ vpg,
    const float* __restrict__ qqg, const float* __restrict__ kkg,
    const float* __restrict__ z,   const float* __restrict__ R,
    const float* __restrict__ t,   const float* __restrict__ hwts,
    const float* __restrict__ Wb,  float* __restrict__ cat) {
  extern __shared__ float sm[];
  float* zs   = sm;                  // 512*132
  float* lg   = zs + N_ * ZP_;       // 16*512 (rows 12..15 zero-padded)
  float* Wbs  = lg + 16 * N_;        // 128*16 (cols 12..15 zero)
  float* qi   = Wbs + Z_ * 16;       // 192
  float* qpi  = qi + H_ * C_;        // 144
  float* qqi  = qpi + H_ * PQ_ * 3;  // 12
  float* gam  = qqi + H_;            // 12
  float* Ri   = gam + H_;            // 9
  float* ti   = Ri + 9;              // 3
  float* optg = ti + 3;              // 288

  const int bi = blockIdx.x;         // b*N + i
  const int tid = threadIdx.x;
  const int lane = tid & 31, lr = lane & 15, kh = lane >> 4;
  const int bb = bi >> 9;            // batch index
  const size_t rowbase = (size_t)bb * N_;

  // ---- stage z[b,i,:,:] HBM -> LDS once (TDM / async / fallback) ---------
  const float* zsrc = z + (size_t)bi * N_ * Z_;
#if IPA_HAS_TDM
  // One Tensor Data Mover descriptor moves the whole 512x128 f32 tile.
  // D# pad fields insert 4 DWORDs of padding every 128 DWORDs -> 132-float
  // LDS row stride (bank-conflict-free column reads for the bias GEMM).
  if (tid < 32) {  // wave 0 issues the single TDM op (EXEC ignored by TDM)
    unsigned long long ga = (unsigned long long)(size_t)zsrc;
    unsigned int ldsoff = (unsigned int)(size_t)zs;  // LDS byte offset
    u32x4 g0;
    g0.x = 1u;                                       // count=1, user D#
    g0.y = ldsoff;                                   // lds_addr
    g0.z = (unsigned int)(ga & 0xffffffffu);         // global_addr[31:0]
    g0.w = (unsigned int)((ga >> 32) & 0x1ffffffu) | (2u << 30);  // type=2
    i32x8 g1;
    g1[0] = (int)((2u << 16)      // data_size = 4 bytes
                  | (1u << 20)    // pad_enable
                  | (6u << 22)    // pad_interval: every 128 DWORDs
                  | (3u << 25));  // pad_amount: 4 DWORDs
    g1[1] = (int)(128u << 16);    // tensor_dim0 = 128 (low 16 of 32b field)
    g1[2] = (int)(512u << 16);    // tensor_dim0 hi=0 | tensor_dim1 = 512
    g1[3] = (int)(128u << 16);    // tensor_dim1 hi=0 | tile_dim0 = 128
    g1[4] = (int)(512u);          // tile_dim1 = 512 | tile_dim2 = 0
    g1[5] = (int)(128u);          // tensor_dim0_stride = 128
    g1[6] = 0;
    g1[7] = 0;
    i32x4 gz4 = {0, 0, 0, 0};
    i32x8 gz8 = {0, 0, 0, 0, 0, 0, 0, 0};
    // 6-arg form (clang-23 / therock-10.0): (g0, g1, g2, g3, g4, cpol)
    __builtin_amdgcn_tensor_load_to_lds(g0, g1, gz4, gz4, gz8, 0);
    __builtin_amdgcn_s_wait_tensorcnt(0);
  }
#elif IPA_HAS_ASYNC
  {
    typedef const __attribute__((address_space(1))) void* gp_t;
    typedef __attribute__((address_space(3))) void* lp_t;
    const float4* zg4 = (const float4*)zsrc;
    float4* zs4 = (float4*)zs;
    for (int idx = tid; idx < N_ * (Z_ / 4); idx += 256) {
      int row = idx >> 5, c4 = idx & 31;  // 32 float4 per z row
      __builtin_amdgcn_global_load_async_to_lds_b128(
          (gp_t)(zg4 + idx), (lp_t)(zs4 + row * (ZP_ / 4) + c4), 0, 0);
    }
#if __has_builtin(__builtin_amdgcn_s_wait_asynccnt)
    __builtin_amdgcn_s_wait_asynccnt(0);
#else
    asm volatile("s_wait_asynccnt 0x0" ::: "memory");
#endif
  }
#else
  {
    const float4* zg4 = (const float4*)zsrc;
    float4* zs4 = (float4*)zs;
    for (int idx = tid; idx < N_ * (Z_ / 4); idx += 256) {
      int row = idx >> 5, c4 = idx & 31;
      zs4[row * (ZP_ / 4) + c4] = zg4[idx];
    }
  }
#endif

  // ---- per-row operands staged while the z DMA is in flight --------------
  for (int idx = tid; idx < Z_ * 16; idx += 256) {
    int zc = idx >> 4, h = idx & 15;
    Wbs[idx] = (h < H_) ? Wb[zc * H_ + h] : 0.f;
  }
  for (int idx = tid; idx < 4 * N_; idx += 256) lg[H_ * N_ + idx] = 0.f;
  if (tid < H_ * C_) qi[tid] = q[(size_t)bi * H_ * C_ + tid];
  if (tid < H_ * PQ_ * 3) qpi[tid] = qpg[(size_t)bi * H_ * PQ_ * 3 + tid];
  if (tid < H_) {
    qqi[tid] = qqg[(size_t)bi * H_ + tid];
    gam[tid] = log1pf(__expf(hwts[tid]));   // softplus
  }
  if (tid < 9) Ri[tid] = R[(size_t)bi * 9 + tid];
  if (tid < 3) ti[tid] = t[(size_t)bi * 3 + tid];
  __syncthreads();

  // ---- bias GEMM from LDS: lg[h,j] = sum_z zs[j,z] * Wbs[z,h] ------------
  for (int jt = (tid >> 5); jt < N_ / 16; jt += 8) {
    v8f acc = {};
    const float* arow = zs + (size_t)(jt * 16 + lr) * ZP_ + 2 * kh;
    for (int k0 = 0; k0 < Z_; k0 += 4) {
      v2f a, b;
      a.x = arow[k0];
      a.y = arow[k0 + 1];
      b.x = Wbs[(k0 + 2 * kh) * 16 + lr];
      b.y = Wbs[(k0 + 2 * kh + 1) * 16 + lr];
      acc = wmma4(a, b, acc);
    }
    if (lr < H_) {
      int jb = jt * 16 + 8 * kh;
      for (int r = 0; r < 8; ++r) lg[lr * N_ + jb + r] = acc[r];
    }
  }
  __syncthreads();

  // ---- logits: wL*(scal + bias) - (wL*wC/2)*gamma*dist2 ------------------
  const float wL = 0.5773502691896258f;   // sqrt(1/3)
  const float wC = 0.23570226039551584f;  // sqrt(2/(9*PQ)) = sqrt(1/18)
  for (int idx = tid; idx < H_ * N_; idx += 256) {
    int h = idx / N_, j = idx - h * N_;
    size_t rj = rowbase + j;
    const float* kj = k + rj * (H_ * C_) + h * C_;
    float sc = 0.f;
    for (int c = 0; c < C_; ++c) sc += qi[h * C_ + c] * kj[c];
    sc *= 0.25f;  // 1/sqrt(C)
    const float* kpj = kpg + rj * (H_ * PQ_ * 3) + h * PQ_ * 3;
    float qk = 0.f;
    for (int e = 0; e < PQ_ * 3; ++e) qk += qpi[h * PQ_ * 3 + e] * kpj[e];
    float d2 = qqi[h] + kkg[rj * H_ + h] - 2.f * qk;
    lg[idx] = wL * (sc + lg[idx]) - (0.5f * wL * wC) * gam[h] * d2;
  }
  __syncthreads();

  // ---- softmax over j per head (one wave32 per row) ----------------------
  for (int h = tid >> 5; h < H_; h += 8) {
    float m = -1e30f;
    for (int j = lane; j < N_; j += 32) m = fmaxf(m, lg[h * N_ + j]);
    for (int o = 16; o; o >>= 1) m = fmaxf(m, __shfl_xor(m, o, 32));
    float ssum = 0.f;
    for (int j = lane; j < N_; j += 32) {
      float e = __expf(lg[h * N_ + j] - m);
      lg[h * N_ + j] = e;
      ssum += e;
    }
    for (int o = 16; o; o >>= 1) ssum += __shfl_xor(ssum, o, 32);
    float inv = 1.f / ssum;
    for (int j = lane; j < N_; j += 32) lg[h * N_ + j] *= inv;
  }
  __syncthreads();

  // ---- o = a @ v (L2-resident) and optg = a @ vp_global ------------------
  float* crow = cat + (size_t)bi * CATW_;
  for (int idx = tid; idx < H_ * C_; idx += 256) {
    int h = idx >> 4;
    const float* vcol = v + rowbase * (H_ * C_) + idx;
    const float* ar = lg + h * N_;
    float acc = 0.f;
    for (int j = 0; j < N_; ++j) acc += ar[j] * vcol[(size_t)j * (H_ * C_)];
    crow[idx] = acc;
  }
  for (int idx = tid; idx < H_ * PV_ * 3; idx += 256) {
    int h = idx / (PV_ * 3);
    const float* vcol = vpg + rowbase * (H_ * PV_ * 3) + idx;
    const float* ar = lg + h * N_;
    float acc = 0.f;
    for (int j = 0; j < N_; ++j) acc += ar[j] * vcol[(size_t)j * (H_ * PV_ * 3)];
    optg[idx] = acc;
  }
  __syncthreads();

  // ---- back to local frame: opt = R^T (optg - t); norms ------------------
  for (int idx = tid; idx < H_ * PV_; idx += 256) {
    float d0 = optg[idx * 3 + 0] - ti[0];
    float d1 = optg[idx * 3 + 1] - ti[1];
    float d2 = optg[idx * 3 + 2] - ti[2];
    float x0 = Ri[0] * d0 + Ri[3] * d1 + Ri[6] * d2;
    float x1 = Ri[1] * d0 + Ri[4] * d1 + Ri[7] * d2;
    float x2 = Ri[2] * d0 + Ri[5] * d1 + Ri[8] * d2;
    crow[H_ * C_ + idx * 3 + 0] = x0;
    crow[H_ * C_ + idx * 3 + 1] = x1;
    crow[H_ * C_ + idx * 3 + 2] = x2;
    crow[H_ * C_ + H_ * PV_ * 3 + idx] = sqrtf(x0 * x0 + x1 * x1 + x2 * x2 + 1e-8f);
  }

  // ---- opair GEMM from LDS: [16(h) x 128(z)] = a[16x512] @ zs[512x128] ---
  {
    int nt = tid >> 5;  // one z-column tile of 16 per wave
    v8f acc = {};
    const float* arow = lg + (size_t)lr * N_ + 2 * kh;   // rows 12..15 zero
    for (int k0 = 0; k0 < N_; k0 += 4) {
      v2f a, b;
      a.x = arow[k0];
      a.y = arow[k0 + 1];
      b.x = zs[(size_t)(k0 + 2 * kh) * ZP_ + nt * 16 + lr];
      b.y = zs[(size_t)(k0 + 2 * kh + 1) * ZP_ + nt * 16 + lr];
      acc = wmma4(a, b, acc);
    }
    for (int r = 0; r < 8; ++r) {
      int h = r + 8 * kh;
      if (h < H_)
        crow[H_ * C_ + 4 * H_ * PV_ + h * Z_ + nt * 16 + lr] = acc[r];
    }
  }
}

// ---------------------------------------------------------------------------
extern "C" void kernel_launch(void* const* d_in, const int* in_sizes, int n_in,
                              void* d_out, int out_size, void* d_ws, size_t ws_size,
                              hipStream_t stream) {
  const float* s    = (const float*)d_in[0];
  const float* z    = (const float*)d_in[1];
  const float* R    = (const float*)d_in[2];
  const float* t    = (const float*)d_in[3];
  const float* Wq   = (const float*)d_in[4];
  const float* Wk   = (const float*)d_in[5];
  const float* Wv   = (const float*)d_in[6];
  const float* Wqp  = (const float*)d_in[7];
  const float* Wkp  = (const float*)d_in[8];
  const float* Wvp  = (const float*)d_in[9];
  const float* Wb   = (const float*)d_in[10];
  const float* hwts = (const float*)d_in[11];
  const float* Wout = (const float*)d_in[12];
  float* out = (float*)d_out;

  const int BN = 2 * 512;  // 1024 rows
  float* ws = (float*)d_ws;
  float* q  = ws;                 // 1024*192
  float* k  = q  + BN * 192;
  float* v  = k  + BN * 192;
  float* qp = v  + BN * 192;      // 1024*144 (local -> global in place)
  float* kp = qp + BN * 144;
  float* vp = kp + BN * 144;      // 1024*288
  float* qq = vp + BN * 288;      // 1024*12
  float* kk = qq + BN * 12;
  float* cat = kk + BN * 12;      // 1024*2112

  // ---- projections: s[1024,384] @ W -> q/k/v/qp/kp/vp (f32 WMMA) ---------
  gemm_f32_wmma<<<dim3(BN / 16, 192 / 16), 32, 0, stream>>>(s, Wq,  q,  BN, 192, 384);
  gemm_f32_wmma<<<dim3(BN / 16, 192 / 16), 32, 0, stream>>>(s, Wk,  k,  BN, 192, 384);
  gemm_f32_wmma<<<dim3(BN / 16, 192 / 16), 32, 0, stream>>>(s, Wv,  v,  BN, 192, 384);
  gemm_f32_wmma<<<dim3(BN / 16, 144 / 16), 32, 0, stream>>>(s, Wqp, qp, BN, 144, 384);
  gemm_f32_wmma<<<dim3(BN / 16, 144 / 16), 32, 0, stream>>>(s, Wkp, kp, BN, 144, 384);
  gemm_f32_wmma<<<dim3(BN / 16, 288 / 16), 32, 0, stream>>>(s, Wvp, vp, BN, 288, 384);

  // ---- frame transforms + squared norms ----------------------------------
  point_to_global<<<(BN * 48 + 255) / 256, 256, 0, stream>>>(qp, R, t, 48, BN * 48);
  point_to_global<<<(BN * 48 + 255) / 256, 256, 0, stream>>>(kp, R, t, 48, BN * 48);
  point_to_global<<<(BN * 96 + 255) / 256, 256, 0, stream>>>(vp, R, t, 96, BN * 96);
  point_sqsum<<<(BN * 12 + 255) / 256, 256, 0, stream>>>(qp, qq, PQ_, BN * 12);
  point_sqsum<<<(BN * 12 + 255) / 256, 256, 0, stream>>>(kp, kk, PQ_, BN * 12);

  // ---- fused attention: z staged once into 320KB LDS ---------------------
  size_t shbytes = (size_t)(N_ * ZP_ + 16 * N_ + Z_ * 16 + 192 + 144 + 12 + 12 +
                            9 + 3 + 288) * sizeof(float);  // ~306 KB < 320 KB
  ipa_attn<<<BN, 256, shbytes, stream>>>(q, k, v, qp, kp, vp, qq, kk, z, R, t,
                                         hwts, Wb, cat);

  // ---- output projection: cat[1024,2112] @ Wout[2112,384] ----------------
  gemm_f32_wmma<<<dim3(BN / 16, 384 / 16), 32, 0, stream>>>(cat, Wout, out,
                                                            BN, 384, 2112);
}